// CausalSelfAttention_57904749084811
// MI455X (gfx1250) — compile-verified
//
#include <hip/hip_runtime.h>
#include <hip/hip_bf16.h>

// ---------------------------------------------------------------------------
// CausalSelfAttention forward, fp32, built on V_WMMA_F32_16X16X4_F32 (CDNA5).
//   B=2, T=2048, C=768, H=12, Dh=64
//   Pipeline: QKV GEMM (32x64 tiles/wave) -> flash attention with async
//   V-tile staging to LDS (double buffered) -> out projection GEMM.
// ---------------------------------------------------------------------------

#define BB   2
#define TT   2048
#define CC   768
#define HH   12
#define DH   64

typedef __attribute__((ext_vector_type(2))) float v2f;
typedef __attribute__((ext_vector_type(8))) float v8f;

// Optional CDNA5 async global->LDS path (guarded: falls back cleanly).
#if defined(__has_builtin)
#  if __has_builtin(__builtin_amdgcn_global_load_async_to_lds_b128) && \
      __has_builtin(__builtin_amdgcn_s_wait_asynccnt)
#    define USE_ASYNC_LDS 1
#  endif
#endif
#ifndef USE_ASYNC_LDS
#  define USE_ASYNC_LDS 0
#endif

#if USE_ASYNC_LDS
// b128 async copy moves one 16-byte vector per lane: params are
// (v4i addrspace(1)* src, v4i addrspace(3)* dst, imm offset, imm cpol).
typedef int v4i __attribute__((__vector_size__(4 * sizeof(int))));
typedef __attribute__((address_space(1))) v4i* as1_v4ip;
typedef __attribute__((address_space(3))) v4i* as3_v4ip;
#endif

// D = A(16x4) * B(4x16) + C, fp32, wave32.
__device__ __forceinline__ v8f wmma4(v2f a, v2f b, v8f c) {
  return __builtin_amdgcn_wmma_f32_16x16x4_f32(
      /*neg_a=*/false, a, /*neg_b=*/false, b,
      /*c_mod=*/(short)0, c, /*reuse_a=*/false, /*reuse_b=*/false);
}

// ---------------------------------------------------------------------------
// Kernel 1: qkv = x @ W_qkv^T, scattered into Q/K/V workspaces [B,H,T,Dh].
//   M = 4096, N = 2304, K = 768.  Each wave: 32(M) x 64(N) tile,
//   8 WMMAs per 6 fragment loads per K-step.
//   grid = (16, 36), block = 256 (8 waves).
// ---------------------------------------------------------------------------
__global__ __launch_bounds__(256) void qkv_gemm_kernel(
    const float* __restrict__ x, const float* __restrict__ Wqkv,
    float* __restrict__ q, float* __restrict__ k, float* __restrict__ v) {
  const int lane = threadIdx.x & 31;
  const int wave = threadIdx.x >> 5;
  const int mbase = (blockIdx.x * 8 + wave) * 32;   // 0..4064
  const int nbase = blockIdx.y * 64;                // 0..2240
  const int nl  = lane & 15;
  const int hi  = lane >> 4;
  const int klo = hi * 2;

  const float* arow0 = x + (size_t)(mbase + nl) * CC + klo;
  const float* arow1 = arow0 + (size_t)16 * CC;
  const float* brow0 = Wqkv + (size_t)(nbase +  0 + nl) * CC + klo;
  const float* brow1 = Wqkv + (size_t)(nbase + 16 + nl) * CC + klo;
  const float* brow2 = Wqkv + (size_t)(nbase + 32 + nl) * CC + klo;
  const float* brow3 = Wqkv + (size_t)(nbase + 48 + nl) * CC + klo;

  v8f acc[2][4];
#pragma unroll
  for (int i = 0; i < 2; ++i)
#pragma unroll
    for (int j = 0; j < 4; ++j) acc[i][j] = (v8f){};

  for (int k0 = 0; k0 < CC; k0 += 64) {
    if (k0 + 64 < CC) {   // prefetch next K-block -> global_prefetch_b8
      __builtin_prefetch(arow0 + k0 + 64, 0, 3);
      __builtin_prefetch(arow1 + k0 + 64, 0, 3);
      __builtin_prefetch(brow0 + k0 + 64, 0, 3);
      __builtin_prefetch(brow1 + k0 + 64, 0, 3);
      __builtin_prefetch(brow2 + k0 + 64, 0, 3);
      __builtin_prefetch(brow3 + k0 + 64, 0, 3);
    }
#pragma unroll
    for (int kk = 0; kk < 64; kk += 4) {
      const int kq = k0 + kk;
      v2f a0 = *reinterpret_cast<const v2f*>(arow0 + kq);
      v2f a1 = *reinterpret_cast<const v2f*>(arow1 + kq);
      v2f b0 = *reinterpret_cast<const v2f*>(brow0 + kq);
      v2f b1 = *reinterpret_cast<const v2f*>(brow1 + kq);
      v2f b2 = *reinterpret_cast<const v2f*>(brow2 + kq);
      v2f b3 = *reinterpret_cast<const v2f*>(brow3 + kq);
      acc[0][0] = wmma4(a0, b0, acc[0][0]);
      acc[0][1] = wmma4(a0, b1, acc[0][1]);
      acc[0][2] = wmma4(a0, b2, acc[0][2]);
      acc[0][3] = wmma4(a0, b3, acc[0][3]);
      acc[1][0] = wmma4(a1, b0, acc[1][0]);
      acc[1][1] = wmma4(a1, b1, acc[1][1]);
      acc[1][2] = wmma4(a1, b2, acc[1][2]);
      acc[1][3] = wmma4(a1, b3, acc[1][3]);
    }
  }

  // Scatter D tiles into head-major Q/K/V.
#pragma unroll
  for (int j = 0; j < 4; ++j) {
    const int n     = nbase + j * 16 + nl;
    const int which = n / CC;          // 0=q 1=k 2=v (64-blocks never straddle)
    const int c     = n % CC;
    const int h     = c >> 6;
    const int d     = c & 63;
    float* dst = (which == 0) ? q : (which == 1) ? k : v;
#pragma unroll
    for (int i = 0; i < 2; ++i) {
#pragma unroll
      for (int r = 0; r < 8; ++r) {
        const int m = mbase + i * 16 + r + (hi << 3);
        const int b = m >> 11;
        const int t = m & (TT - 1);
        dst[(((size_t)(b * HH + h) * TT + t) << 6) + d] = acc[i][j][r];
      }
    }
  }
}

// ---------------------------------------------------------------------------
// Kernel 2: flash attention, one wave per (b, h, 16-row q-tile).
//   S = (Q*scale) K^T via 16 WMMAs; online softmax in C/D layout;
//   P relaid via LDS; V tile async-staged to LDS (double buffered);
//   O += P V via 16 WMMAs per k-tile.
// ---------------------------------------------------------------------------
__global__ __launch_bounds__(32) void attn_kernel(
    const float* __restrict__ Q, const float* __restrict__ K,
    const float* __restrict__ V, float* __restrict__ O) {
  __shared__ float Pt[16 * 16];
#if USE_ASYNC_LDS
  __shared__ float Vt[2][16 * DH];   // 2 x 4KB double buffer
#endif

  const int lane = threadIdx.x & 31;
  const int qt = blockIdx.x;      // q-tile 0..127
  const int h  = blockIdx.y;
  const int b  = blockIdx.z;
  const int nl  = lane & 15;
  const int hi  = lane >> 4;
  const int klo = hi * 2;

  const size_t headBase = ((size_t)(b * HH + h) * TT) << 6;   // * DH
  const float* Qh = Q + headBase;
  const float* Kh = K + headBase;
  const float* Vh = V + headBase;

  const float scale = 0.125f;     // 1/sqrt(64), folded into Q fragments

  // Preload Q tile (16x64) as 16 A-fragments, pre-scaled.
  v2f aq[16];
  {
    const float* qrow = Qh + (size_t)(qt * 16 + nl) * DH + klo;
#pragma unroll
    for (int kb = 0; kb < 16; ++kb) {
      v2f t = *reinterpret_cast<const v2f*>(qrow + 4 * kb);
      t[0] *= scale; t[1] *= scale;
      aq[kb] = t;
    }
  }

  v8f acco[4];
#pragma unroll
  for (int t = 0; t < 4; ++t) acco[t] = (v8f){};
  float mrow[8], lsum[8];
#pragma unroll
  for (int r = 0; r < 8; ++r) { mrow[r] = -__builtin_inff(); lsum[r] = 0.0f; }

#if USE_ASYNC_LDS
  // Issue 8x b128 async copies (4KB V tile -> LDS buffer). In-order retire.
  auto issueV = [&](int kt, int buf) {
    const char* src = (const char*)(Vh + ((size_t)(kt * 16) << 6)) + lane * 16;
    char* dst = (char*)&Vt[buf][0] + lane * 16;
#pragma unroll
    for (int c = 0; c < 8; ++c) {
      __builtin_amdgcn_global_load_async_to_lds_b128(
          (as1_v4ip)(src + c * 512), (as3_v4ip)(dst + c * 512), 0, 0);
    }
  };
  issueV(0, 0);
#endif

  for (int kt = 0; kt <= qt; ++kt) {
    // ---- S = Q K^T (16x16) ----
    const float* krow = Kh + (size_t)(kt * 16 + nl) * DH + klo;
    v8f s = {};
#pragma unroll
    for (int kb = 0; kb < 16; ++kb) {
      v2f bk = *reinterpret_cast<const v2f*>(krow + 4 * kb);
      s = wmma4(aq[kb], bk, s);
    }

#if USE_ASYNC_LDS
    if (kt < qt) issueV(kt + 1, (kt + 1) & 1);   // overlap next V copy
#endif

    // ---- causal mask (diagonal tile only) ----
    if (kt == qt) {
#pragma unroll
      for (int r = 0; r < 8; ++r)
        if (nl > r + (hi << 3)) s[r] = -__builtin_inff();
    }

    // ---- online softmax: row stats via 16-lane shuffles ----
    float p[8];
#pragma unroll
    for (int r = 0; r < 8; ++r) {
      float mx = s[r];
      mx = fmaxf(mx, __shfl_xor(mx, 1, 32));
      mx = fmaxf(mx, __shfl_xor(mx, 2, 32));
      mx = fmaxf(mx, __shfl_xor(mx, 4, 32));
      mx = fmaxf(mx, __shfl_xor(mx, 8, 32));
      const float newm = fmaxf(mrow[r], mx);
      const float f    = __expf(mrow[r] - newm);   // 0 on first tile
      const float pv   = __expf(s[r] - newm);
      float ps = pv;
      ps += __shfl_xor(ps, 1, 32);
      ps += __shfl_xor(ps, 2, 32);
      ps += __shfl_xor(ps, 4, 32);
      ps += __shfl_xor(ps, 8, 32);
      lsum[r] = lsum[r] * f + ps;
      mrow[r] = newm;
      p[r] = pv;
      acco[0][r] *= f; acco[1][r] *= f; acco[2][r] *= f; acco[3][r] *= f;
    }

    // ---- relayout P (C/D layout -> A fragments) through LDS ----
    __syncthreads();                       // WAR vs previous tile's reads
#pragma unroll
    for (int r = 0; r < 8; ++r)
      Pt[(r + (hi << 3)) * 16 + nl] = p[r];
    __syncthreads();
    v2f ap[4];
#pragma unroll
    for (int j = 0; j < 4; ++j)
      ap[j] = *reinterpret_cast<const v2f*>(&Pt[nl * 16 + 4 * j + klo]);

    // ---- O += P V  (16x16 @ 16x64 as 4 K-steps x 4 N-tiles) ----
#if USE_ASYNC_LDS
    // Tile kt resident once outstanding count drops to the next tile's 8.
    if (kt < qt) __builtin_amdgcn_s_wait_asynccnt(8);
    else         __builtin_amdgcn_s_wait_asynccnt(0);
    const float* vbase = &Vt[kt & 1][0];
#else
    const float* vbase = Vh + ((size_t)(kt * 16) << 6);
#endif
#pragma unroll
    for (int j = 0; j < 4; ++j) {
      const int kr = 4 * j + klo;
#pragma unroll
      for (int t = 0; t < 4; ++t) {
        const int col = t * 16 + nl;
        v2f bv;
        bv[0] = vbase[(kr << 6) + col];
        bv[1] = vbase[((kr + 1) << 6) + col];
        acco[t] = wmma4(ap[j], bv, acco[t]);
      }
    }
  }

  // ---- normalize, store O in [B,T,C] layout ----
  float* orow = O + ((size_t)b * TT + (size_t)qt * 16) * CC + h * DH;
#pragma unroll
  for (int r = 0; r < 8; ++r) {
    const float inv = 1.0f / lsum[r];
    const int m = r + (hi << 3);
#pragma unroll
    for (int t = 0; t < 4; ++t)
      orow[(size_t)m * CC + t * 16 + nl] = acco[t][r] * inv;
  }
}

// ---------------------------------------------------------------------------
// Kernel 3: out = attn_out @ W_proj^T  (M=4096, N=768, K=768)
//   Each wave: 32(M) x 64(N) tile. grid = (16, 12), block = 256 (8 waves).
// ---------------------------------------------------------------------------
__global__ __launch_bounds__(256) void proj_gemm_kernel(
    const float* __restrict__ a, const float* __restrict__ Wp,
    float* __restrict__ out) {
  const int lane = threadIdx.x & 31;
  const int wave = threadIdx.x >> 5;
  const int mbase = (blockIdx.x * 8 + wave) * 32;
  const int nbase = blockIdx.y * 64;
  const int nl  = lane & 15;
  const int hi  = lane >> 4;
  const int klo = hi * 2;

  const float* arow0 = a + (size_t)(mbase + nl) * CC + klo;
  const float* arow1 = arow0 + (size_t)16 * CC;
  const float* brow0 = Wp + (size_t)(nbase +  0 + nl) * CC + klo;
  const float* brow1 = Wp + (size_t)(nbase + 16 + nl) * CC + klo;
  const float* brow2 = Wp + (size_t)(nbase + 32 + nl) * CC + klo;
  const float* brow3 = Wp + (size_t)(nbase + 48 + nl) * CC + klo;

  v8f acc[2][4];
#pragma unroll
  for (int i = 0; i < 2; ++i)
#pragma unroll
    for (int j = 0; j < 4; ++j) acc[i][j] = (v8f){};

  for (int k0 = 0; k0 < CC; k0 += 64) {
    if (k0 + 64 < CC) {
      __builtin_prefetch(arow0 + k0 + 64, 0, 3);
      __builtin_prefetch(arow1 + k0 + 64, 0, 3);
      __builtin_prefetch(brow0 + k0 + 64, 0, 3);
      __builtin_prefetch(brow1 + k0 + 64, 0, 3);
      __builtin_prefetch(brow2 + k0 + 64, 0, 3);
      __builtin_prefetch(brow3 + k0 + 64, 0, 3);
    }
#pragma unroll
    for (int kk = 0; kk < 64; kk += 4) {
      const int kq = k0 + kk;
      v2f a0 = *reinterpret_cast<const v2f*>(arow0 + kq);
      v2f a1 = *reinterpret_cast<const v2f*>(arow1 + kq);
      v2f b0 = *reinterpret_cast<const v2f*>(brow0 + kq);
      v2f b1 = *reinterpret_cast<const v2f*>(brow1 + kq);
      v2f b2 = *reinterpret_cast<const v2f*>(brow2 + kq);
      v2f b3 = *reinterpret_cast<const v2f*>(brow3 + kq);
      acc[0][0] = wmma4(a0, b0, acc[0][0]);
      acc[0][1] = wmma4(a0, b1, acc[0][1]);
      acc[0][2] = wmma4(a0, b2, acc[0][2]);
      acc[0][3] = wmma4(a0, b3, acc[0][3]);
      acc[1][0] = wmma4(a1, b0, acc[1][0]);
      acc[1][1] = wmma4(a1, b1, acc[1][1]);
      acc[1][2] = wmma4(a1, b2, acc[1][2]);
      acc[1][3] = wmma4(a1, b3, acc[1][3]);
    }
  }

#pragma unroll
  for (int j = 0; j < 4; ++j) {
    const int n = nbase + j * 16 + nl;
#pragma unroll
    for (int i = 0; i < 2; ++i) {
#pragma unroll
      for (int r = 0; r < 8; ++r) {
        const int m = mbase + i * 16 + r + (hi << 3);
        out[(size_t)m * CC + n] = acc[i][j][r];
      }
    }
  }
}

// ---------------------------------------------------------------------------
// Host launcher
// ---------------------------------------------------------------------------
extern "C" void kernel_launch(void* const* d_in, const int* in_sizes, int n_in,
                              void* d_out, int out_size, void* d_ws, size_t ws_size,
                              hipStream_t stream) {
  const float* x     = (const float*)d_in[0];   // [B,T,C]
  const float* Wqkv  = (const float*)d_in[1];   // [3C,C]
  const float* Wproj = (const float*)d_in[2];   // [C,C]
  float* out = (float*)d_out;                   // [B,T,C]

  const size_t headElems = (size_t)BB * HH * TT * DH;   // 3,145,728 floats
  float* wsQ = (float*)d_ws;
  float* wsK = wsQ + headElems;
  float* wsV = wsK + headElems;
  float* wsO = wsV + headElems;                 // [B,T,C] == same elem count

  // 1) QKV projection: 128 M-supertiles x 36 N-supertiles, 8 waves/block.
  qkv_gemm_kernel<<<dim3(16, 36), 256, 0, stream>>>(x, Wqkv, wsQ, wsK, wsV);

  // 2) Flash attention: one wave per (q-tile, head, batch).
  attn_kernel<<<dim3(TT / 16, HH, BB), 32, 0, stream>>>(wsQ, wsK, wsV, wsO);

  // 3) Output projection: 128 M-supertiles x 12 N-supertiles, 8 waves/block.
  proj_gemm_kernel<<<dim3(16, 12), 256, 0, stream>>>(wsO, Wproj, out);
}